// GNNClassifier_19602230739293
// MI455X (gfx1250) — compile-verified
//
#include <hip/hip_runtime.h>
#include <hip/hip_bf16.h>

// ---------------------------------------------------------------------------
// GNN classifier: 2x GCNConv(relu) -> mean pool -> MLP head.
// GEMMs use CDNA5 V_WMMA_F32_16X16X4_F32 (full fp32 precision, wave32).
// B is staged in LDS pre-paired ([K/2][64] float2) so each WMMA B-fragment
// is a single ds_load_b64 into an aligned VGPR pair (no register shuffling).
// ---------------------------------------------------------------------------

typedef __attribute__((ext_vector_type(2))) float v2f;
typedef __attribute__((ext_vector_type(8))) float v8f;

#define H_DIM 64

// ---------------- generic zero ----------------
__global__ void zero_kernel(float* __restrict__ p, long long n) {
    long long i = (long long)blockIdx.x * blockDim.x + threadIdx.x;
    long long stride = (long long)gridDim.x * blockDim.x;
    for (; i < n; i += stride) p[i] = 0.0f;
}

// ---------------- degree (edges only; +1 self loop added later) ----------------
__global__ void deg_kernel(const long long* __restrict__ ei, float* __restrict__ deg,
                           long long E) {
    long long i = (long long)blockIdx.x * blockDim.x + threadIdx.x;
    if (i < E) {
        int d = (int)ei[E + i];  // dst row of edge_index
        atomicAdd(&deg[d], 1.0f);
    }
}

__global__ void dinv_kernel(const float* __restrict__ deg, float* __restrict__ dinv, int n) {
    int i = blockIdx.x * blockDim.x + threadIdx.x;
    if (i < n) dinv[i] = rsqrtf(deg[i] + 1.0f);  // +1 for the self loop
}

// ---------------- WMMA fp32 GEMM: C[rows,64] = A[rows,K] @ B[K,64] ----------------
// One wave computes a 16x64 output slab using V_WMMA_F32_16X16X4_F32.
// B (<=64KB) is staged in LDS, pre-paired along K, shared by 8 waves.
__global__ __launch_bounds__(256)
void gemm_wmma_kernel(const float* __restrict__ A, const float* __restrict__ B,
                      float* __restrict__ C, int rows, int K) {
    extern __shared__ float sB[];  // K/2 * 64 float2 == K*64 floats
    const int tid = threadIdx.x;
    // Stage B pre-paired: pair p=k>>1 at sB2[p*64+n] = (B[2p][n], B[2p+1][n])
    for (int i = tid; i < K * H_DIM; i += blockDim.x) {
        const int k = i >> 6;
        const int n = i & 63;
        sB[(((k >> 1) * H_DIM) + n) * 2 + (k & 1)] = B[i];
    }
    __syncthreads();

    const int wave = tid >> 5;
    const int lane = tid & 31;
    const int tile = blockIdx.x * (blockDim.x >> 5) + wave;
    const int row0 = tile * 16;
    if (row0 >= rows) return;  // wave-uniform exit

    const int m = lane & 15;       // row within tile (A) / column within 16-tile (B,C)
    const int khalf = lane >> 4;   // which K-pair this half-wave owns

    // A fragment base: lane holds A[row][kb + 2*khalf + {0,1}]
    int arow = row0 + m;
    if (arow >= rows) arow = rows - 1;  // clamp (harmless, stores are guarded)
    const float* aptr = A + (long long)arow * K + 2 * khalf;
    const float2* sB2 = (const float2*)sB;

    v8f acc[4] = {v8f{}, v8f{}, v8f{}, v8f{}};

    for (int kb = 0; kb < K; kb += 4) {
        const float2 av = *(const float2*)(aptr + kb);
        v2f a;
        a.x = av.x;
        a.y = av.y;
        // pair row for this lane: (kb + 2*khalf) >> 1 == kb/2 + khalf
        const int pr = ((kb >> 1) + khalf) * H_DIM + m;
#pragma unroll
        for (int nt = 0; nt < 4; ++nt) {
            const float2 bv = sB2[pr + nt * 16];  // single ds_load_b64
            v2f b;
            b.x = bv.x;
            b.y = bv.y;
            acc[nt] = __builtin_amdgcn_wmma_f32_16x16x4_f32(
                false, a, false, b, (short)0, acc[nt], false, false);
        }
    }

    // C layout: VGPR v, lane -> row = v + 8*khalf, col = nt*16 + m
#pragma unroll
    for (int nt = 0; nt < 4; ++nt) {
#pragma unroll
        for (int v = 0; v < 8; ++v) {
            int r = row0 + v + 8 * khalf;
            if (r < rows) C[(long long)r * H_DIM + nt * 16 + m] = acc[nt][v];
        }
    }
}

// ---------------- edge scatter: agg[dst] += h[src] * dinv[src]*dinv[dst] ----------------
__global__ void scatter_kernel(const float* __restrict__ h, const long long* __restrict__ ei,
                               const float* __restrict__ dinv, float* __restrict__ agg,
                               long long E) {
    const long long total = E * H_DIM;
    long long idx = (long long)blockIdx.x * blockDim.x + threadIdx.x;
    const long long stride = (long long)gridDim.x * blockDim.x;
    for (; idx < total; idx += stride) {
        const long long e = idx >> 6;
        const int f = (int)(idx & 63);
        const int s = (int)ei[e];
        const int d = (int)ei[E + e];
        const float nrm = dinv[s] * dinv[d];
        atomicAdd(&agg[(long long)d * H_DIM + f], h[(long long)s * H_DIM + f] * nrm);
    }
}

// ---------------- out = relu(agg + h*dinv^2 (self loop) + bias) ----------------
__global__ void finalize_kernel(const float* __restrict__ agg, const float* __restrict__ h,
                                const float* __restrict__ dinv, const float* __restrict__ bias,
                                float* __restrict__ out, int n) {
    long long idx = (long long)blockIdx.x * blockDim.x + threadIdx.x;
    const long long total = (long long)n * H_DIM;
    if (idx < total) {
        const int i = (int)(idx >> 6);
        const int f = (int)(idx & 63);
        const float di = dinv[i];
        const float v = agg[idx] + h[idx] * di * di + bias[f];
        out[idx] = fmaxf(v, 0.0f);
    }
}

// ---------------- mean-pool accumulation ----------------
__global__ void pool_kernel(const float* __restrict__ x, const long long* __restrict__ batch,
                            float* __restrict__ pooled, float* __restrict__ counts, int n) {
    long long idx = (long long)blockIdx.x * blockDim.x + threadIdx.x;
    const long long total = (long long)n * H_DIM;
    if (idx < total) {
        const int i = (int)(idx >> 6);
        const int f = (int)(idx & 63);
        const int g = (int)batch[i];
        atomicAdd(&pooled[g * H_DIM + f], x[idx]);
        if (f == 0) atomicAdd(&counts[g], 1.0f);
    }
}

// ---------------- MLP head (4 graphs -> [4,4]); tiny, single block ----------------
__global__ __launch_bounds__(128)
void head_kernel(const float* __restrict__ pooled, const float* __restrict__ counts,
                 const float* __restrict__ fc1_w, const float* __restrict__ fc1_b,
                 const float* __restrict__ fc2_w, const float* __restrict__ fc2_b,
                 float* __restrict__ out) {
    __shared__ float sp[4 * H_DIM];
    __shared__ float sz[4 * 128];
    const int tid = threadIdx.x;
    for (int i = tid; i < 4 * H_DIM; i += 128) {
        sp[i] = pooled[i] / fmaxf(counts[i >> 6], 1.0f);
    }
    __syncthreads();
    {
        const int j = tid;  // 0..127
        for (int g = 0; g < 4; ++g) {
            float s = fc1_b[j];
            for (int k = 0; k < H_DIM; ++k) s += sp[g * H_DIM + k] * fc1_w[k * 128 + j];
            sz[g * 128 + j] = fmaxf(s, 0.0f);
        }
    }
    __syncthreads();
    if (tid < 16) {
        const int g = tid >> 2, c = tid & 3;
        float s = fc2_b[c];
        for (int k = 0; k < 128; ++k) s += sz[g * 128 + k] * fc2_w[k * 4 + c];
        out[g * 4 + c] = s;
    }
}

// ---------------------------------------------------------------------------
extern "C" void kernel_launch(void* const* d_in, const int* in_sizes, int n_in,
                              void* d_out, int out_size, void* d_ws, size_t ws_size,
                              hipStream_t stream) {
    const float*     x     = (const float*)d_in[0];
    const long long* ei    = (const long long*)d_in[1];   // [2,E] int64
    const long long* batch = (const long long*)d_in[2];   // [N] int64
    const float*     W1    = (const float*)d_in[3];
    const float*     b1    = (const float*)d_in[4];
    const float*     W2    = (const float*)d_in[5];
    const float*     b2    = (const float*)d_in[6];
    const float*     fc1_w = (const float*)d_in[7];
    const float*     fc1_b = (const float*)d_in[8];
    const float*     fc2_w = (const float*)d_in[9];
    const float*     fc2_b = (const float*)d_in[10];
    float* out = (float*)d_out;

    const int       F_IN = 256;
    const long long N    = in_sizes[0] / F_IN;
    const long long E    = in_sizes[1] / 2;

    // workspace layout (floats)
    float* ws      = (float*)d_ws;
    float* buf_h   = ws;                     // [N,64] GEMM output (h)
    float* buf_agg = buf_h + N * H_DIM;      // [N,64] scatter accumulator
    float* buf_out = buf_agg + N * H_DIM;    // [N,64] post-relu activation
    float* deg     = buf_out + N * H_DIM;    // [N]
    float* dinv    = deg + N;                // [N]
    float* pooled  = dinv + N;               // [4,64]
    float* counts  = pooled + 4 * H_DIM;     // [4]

    const long long NH = N * H_DIM;
    const int ZB = 1024;  // blocks for grid-stride zero

    // norms
    zero_kernel<<<ZB, 256, 0, stream>>>(deg, N);
    zero_kernel<<<64, 256, 0, stream>>>(pooled, 4 * H_DIM + 4);
    deg_kernel<<<(int)((E + 255) / 256), 256, 0, stream>>>(ei, deg, E);
    dinv_kernel<<<(int)((N + 255) / 256), 256, 0, stream>>>(deg, dinv, (int)N);

    const int tiles      = (int)((N + 15) / 16);
    const int gemm_grid  = (tiles + 7) / 8;           // 8 waves / block
    const long long nhB  = (NH + 255) / 256;

    // ---- layer 1 ----
    gemm_wmma_kernel<<<gemm_grid, 256, F_IN * H_DIM * sizeof(float), stream>>>(
        x, W1, buf_h, (int)N, F_IN);
    zero_kernel<<<ZB, 256, 0, stream>>>(buf_agg, NH);
    scatter_kernel<<<8192, 256, 0, stream>>>(buf_h, ei, dinv, buf_agg, E);
    finalize_kernel<<<(int)nhB, 256, 0, stream>>>(buf_agg, buf_h, dinv, b1, buf_out, (int)N);

    // ---- layer 2 ----
    gemm_wmma_kernel<<<gemm_grid, 256, H_DIM * H_DIM * sizeof(float), stream>>>(
        buf_out, W2, buf_h, (int)N, H_DIM);
    zero_kernel<<<ZB, 256, 0, stream>>>(buf_agg, NH);
    scatter_kernel<<<8192, 256, 0, stream>>>(buf_h, ei, dinv, buf_agg, E);
    finalize_kernel<<<(int)nhB, 256, 0, stream>>>(buf_agg, buf_h, dinv, b2, buf_out, (int)N);

    // ---- pool + head ----
    pool_kernel<<<(int)nhB, 256, 0, stream>>>(buf_out, batch, pooled, counts, (int)N);
    head_kernel<<<1, 128, 0, stream>>>(pooled, counts, fc1_w, fc1_b, fc2_w, fc2_b, out);
}